// GMP_23545010717400
// MI455X (gfx1250) — compile-verified
//
#include <hip/hip_runtime.h>

// GMP (generalized memory polynomial) via CDNA5 WMMA f32 16x16x4.
//
// y[b,t] = sum_m c[b,t+m-10] * ( w_lin[m] + sum_{d,i} |c[b,t+m+i-10]|^(d+1) * w_cross[d,i,m] )
//
// Reformulated as G = Q x W2 with K = 84 (21 envelope positions s=m+i, 4 powers each),
// executed as 21 V_WMMA_F32_16X16X4_F32 per 16-sample tile (K-chunk kc == s, the 4
// K-slots == powers a^1..a^4). Full fp32 precision throughout.

typedef __attribute__((ext_vector_type(2))) float v2f;
typedef __attribute__((ext_vector_type(8))) float v8f;

#define T_LEN   8192
#define M_TAPS  11
#define BLOCK_T 256                       // time samples per workgroup (8 wave32)
#define WIN     (BLOCK_T + 2*(M_TAPS-1))  // staged c window: [T0-10, T0+266) -> 276
#define NCHUNK  21                        // envelope positions s = 0..20 (K = 84 = 21*4)

__global__ __launch_bounds__(256)
void gmp_wmma_kernel(const float* __restrict__ x,   // (B, T, 2) interleaved re/im
                     const float* __restrict__ w,   // (495,) = 11 linear + 4*11*11 cross
                     float*       __restrict__ out) // (B, T, 2)
{
    __shared__ float s_re[WIN];
    __shared__ float s_im[WIN];
    __shared__ float s_a [WIN];
    __shared__ float s_g [8][32][17];     // per-wave G tiles, padded (17 odd -> no bank conflicts)

    const int b   = blockIdx.y;
    const int T0  = blockIdx.x * BLOCK_T;
    const int tid = threadIdx.x;

    // ---- Stage c window and envelope |c| into LDS (zero-fill outside [0,T)) ----
    const float2* __restrict__ x2 = (const float2*)x;
    for (int i = tid; i < WIN; i += 256) {
        const int t = T0 - (M_TAPS - 1) + i;
        float re = 0.f, im = 0.f;
        if (t >= 0 && t < T_LEN) {
            const float2 cv = x2[b * T_LEN + t];
            re = cv.x; im = cv.y;
        }
        s_re[i] = re;
        s_im[i] = im;
        s_a [i] = sqrtf(re * re + im * im);
    }
    __syncthreads();

    const int lane = tid & 31;
    const int wid  = tid >> 5;           // wave id 0..7; wave owns samples [T0+32*wid, +32)
    const int mrow = lane & 15;
    const bool hi  = (lane >= 16);       // hi half-wave: K-slots 2,3 -> powers a^3, a^4

    // ---- Precompute B operands: B[kc] = W2 rows (4 x 16), band w_cross[d, kc-m, m] ----
    // Assumed fp32 4x16 B layout (mirror of documented 16x4 A layout):
    //   lanes 0-15: col N=lane, v0=row K0, v1=row K1; lanes 16-31: v0=K2, v1=K3.
    v2f Bv[NCHUNK];
    const int dbase = hi ? 2 : 0;
#pragma unroll
    for (int kc = 0; kc < NCHUNK; ++kc) {
        const int i0 = kc - mrow;                     // cross-memory index i
        float b0 = 0.f, b1 = 0.f;
        if (mrow < M_TAPS && i0 >= 0 && i0 < M_TAPS) {
            b0 = w[M_TAPS + (dbase + 0) * (M_TAPS * M_TAPS) + i0 * M_TAPS + mrow];
            b1 = w[M_TAPS + (dbase + 1) * (M_TAPS * M_TAPS) + i0 * M_TAPS + mrow];
        }
        Bv[kc].x = b0;
        Bv[kc].y = b1;
    }

    // ---- G = Q x W2 for two 16-row tiles (rows = consecutive time samples) ----
    v8f acc0 = {0.f, 0.f, 0.f, 0.f, 0.f, 0.f, 0.f, 0.f};
    v8f acc1 = {0.f, 0.f, 0.f, 0.f, 0.f, 0.f, 0.f, 0.f};
    const int base = wid * 32 + mrow;    // LDS index of a_ext[t + 0] for this lane's row
#pragma unroll
    for (int kc = 0; kc < NCHUNK; ++kc) {
        const float a0 = s_a[base + kc];          // tile0 row, envelope pos s=kc
        const float a1 = s_a[base + 16 + kc];     // tile1 row
        const float q0 = a0 * a0;
        const float q1 = a1 * a1;
        v2f A0, A1;
        A0.x = hi ? q0 * a0 : a0;   // K-slot 0/2 : a^1 / a^3
        A0.y = hi ? q0 * q0 : q0;   // K-slot 1/3 : a^2 / a^4
        A1.x = hi ? q1 * a1 : a1;
        A1.y = hi ? q1 * q1 : q1;
        acc0 = __builtin_amdgcn_wmma_f32_16x16x4_f32(false, A0, false, Bv[kc],
                                                     (short)0, acc0, false, false);
        acc1 = __builtin_amdgcn_wmma_f32_16x16x4_f32(false, A1, false, Bv[kc],
                                                     (short)0, acc1, false, false);
    }

    // ---- Spill D tiles to LDS (documented layout: lane = column N, VGPR v = row M
    //      for lanes 0-15, row M+8 for lanes 16-31) ----
    const int rbase = hi ? 8 : 0;
#pragma unroll
    for (int v = 0; v < 8; ++v) {
        s_g[wid][rbase + v     ][mrow] = acc0[v];
        s_g[wid][rbase + v + 16][mrow] = acc1[v];
    }
    __syncthreads();

    // ---- Final 11-tap complex combine: one sample per lane ----
    const int r  = lane;                  // row within this wave's 32-sample block
    const int t  = T0 + wid * 32 + r;
    const int cb = wid * 32 + r;          // LDS index of c[t-10]
    float yr = 0.f, yi = 0.f;
#pragma unroll
    for (int m = 0; m < M_TAPS; ++m) {
        const float g = w[m] + s_g[wid][r][m];    // w[m] -> uniform scalar load
        yr = fmaf(s_re[cb + m], g, yr);
        yi = fmaf(s_im[cb + m], g, yi);
    }
    ((float2*)out)[b * T_LEN + t] = make_float2(yr, yi);
}

extern "C" void kernel_launch(void* const* d_in, const int* in_sizes, int n_in,
                              void* d_out, int out_size, void* d_ws, size_t ws_size,
                              hipStream_t stream) {
    const float* x = (const float*)d_in[0];   // (B, 8192, 2) float32
    const float* w = (const float*)d_in[1];   // (1, 495)    float32
    float* out = (float*)d_out;               // (B, 8192, 2) float32
    const int B = in_sizes[0] / (T_LEN * 2);
    dim3 grid(T_LEN / BLOCK_T, B);
    gmp_wmma_kernel<<<grid, 256, 0, stream>>>(x, w, out);
}